// STFT_80745385164951
// MI455X (gfx1250) — compile-verified
//
#include <hip/hip_runtime.h>

// ---------------------------------------------------------------------------
// STFT as fp32 WMMA GEMM for gfx1250 (MI455X).
//   out_re[b,t,f] = sum_n x_pad[b, t*512+n] * w_re[f,n]   (same for imag)
// A (frames x K) from LDS-staged x chunk (frames overlap 4x -> 4x HBM save).
// B (K x freqs): weights repacked once per launch into WMMA-fragment order in
// d_ws so the hot loop issues fully-coalesced global_load_b128 (8 lines/instr,
// 100% utilized) instead of 16-way divergent b64 gathers.
// Full fp32 precision via V_WMMA_F32_16X16X4_F32 (memory-bound problem:
// ~116 GFLOP vs ~175MB minimum HBM traffic; weights stay L2-resident).
// ---------------------------------------------------------------------------

typedef float v2f __attribute__((ext_vector_type(2)));
typedef float v4f __attribute__((ext_vector_type(4)));
typedef float v8f __attribute__((ext_vector_type(8)));

#define N_FFT        2048
#define HOP          512
#define N_FREQ       1025
#define N_SAMP       441000
#define N_BATCH      16
#define T_FRAMES     862                     // (441000/512)+1 VALID frames
#define PAD          (N_FFT / 2)             // 1024 reflect pad
#define M_TILES      2                       // 16-frame M-tiles per workgroup
#define FRAMES_WG    (16 * M_TILES)          // 32 frames per WG
#define LDS_FLOATS   ((FRAMES_WG - 1) * HOP + N_FFT)   // 17920 floats = 71680 B
#define NTILES       ((N_FREQ + 15) / 16)    // 65 freq tiles (last zero-padded)
#define WAVES_WG     8
#define THREADS      (WAVES_WG * 32)
#define BLK_PER_B    ((T_FRAMES + FRAMES_WG - 1) / FRAMES_WG)   // 27

// Packed-weight geometry: entry (nt, q, lane) is a float4 holding the two
// K-pairs lane `hi` consumes for K sub-groups 8q..8q+7.
#define QGROUPS          (N_FFT / 8)                     // 256
#define PACK_PER_NT      (QGROUPS * 32)                  // 8192 float4
#define PACK_PER_MAT     (NTILES * PACK_PER_NT)          // 532480 float4
#define PACK_BYTES       ((size_t)2 * PACK_PER_MAT * 16) // 17,039,360 B

// ---------------------------------------------------------------------------
// One-time (per launch) weight repack into fragment order; zero-pads f>=1025.
// ---------------------------------------------------------------------------
__global__ void stft_repack_weights(const float* __restrict__ wre,
                                    const float* __restrict__ wim,
                                    v4f* __restrict__ packed) {
    const int tid = blockIdx.x * blockDim.x + threadIdx.x;
    if (tid >= 2 * PACK_PER_MAT) return;
    const int m    = tid / PACK_PER_MAT;        // 0 = real, 1 = imag
    const int r    = tid % PACK_PER_MAT;
    const int nt   = r / PACK_PER_NT;
    const int e    = r % PACK_PER_NT;
    const int q    = e >> 5;
    const int lane = e & 31;
    const int row  = lane & 15;
    const int hi   = lane >> 4;
    const int f    = nt * 16 + row;
    v4f v = {0.f, 0.f, 0.f, 0.f};
    if (f < N_FREQ) {
        const float* w = (m == 0 ? wre : wim) + (long)f * N_FFT;
        const int k = 8 * q + 2 * hi;
        v.x = w[k];     v.y = w[k + 1];
        v.z = w[k + 4]; v.w = w[k + 5];
    }
    packed[tid] = v;
}

// ---------------------------------------------------------------------------
// Shared store helper (C/D layout: VGPR v -> M = v + 8*hi, N = lane&15)
// ---------------------------------------------------------------------------
__device__ __forceinline__
void store_tiles(float* __restrict__ out, long out_half, int b, int t0,
                 int f, int hi, const v8f& re0, const v8f& im0,
                 const v8f& re1, const v8f& im1) {
    if (f >= N_FREQ) return;
#pragma unroll
    for (int v = 0; v < 8; ++v) {
        const int m  = v + hi * 8;
        const int ta = t0 + m;
        if (ta < T_FRAMES) {
            const long idx = ((long)b * T_FRAMES + ta) * N_FREQ + f;
            out[idx]            = re0[v];
            out[out_half + idx] = im0[v];
        }
        const int tb = t0 + 16 + m;
        if (tb < T_FRAMES) {
            const long idx = ((long)b * T_FRAMES + tb) * N_FREQ + f;
            out[idx]            = re1[v];
            out[out_half + idx] = im1[v];
        }
    }
}

// ---------------------------------------------------------------------------
// Main kernel, packed-weight path (coalesced b128 B loads).
// ---------------------------------------------------------------------------
__global__ __launch_bounds__(THREADS, 1)
void stft_wmma_f32_packed(const float* __restrict__ x,
                          const v4f* __restrict__ packed,
                          float* __restrict__ out) {
    extern __shared__ float lds_x[];

    const int wg   = blockIdx.x;
    const int b    = wg / BLK_PER_B;
    const int t0   = (wg % BLK_PER_B) * FRAMES_WG;
    const long j0  = (long)t0 * HOP - PAD;

    const float* xb = x + (long)b * N_SAMP;
    for (int i = threadIdx.x; i < LDS_FLOATS; i += THREADS) {
        long j = j0 + i;
        if (j < 0)            j = -j;
        else if (j >= N_SAMP) j = 2L * (N_SAMP - 1) - j;
        lds_x[i] = xb[j];
    }
    __syncthreads();

    const int lane = threadIdx.x & 31;
    const int wave = threadIdx.x >> 5;
    const int row  = lane & 15;
    const int hi   = lane >> 4;
    const int koff = hi << 1;
    const long OUT_HALF = (long)N_BATCH * T_FRAMES * N_FREQ;

    const v4f* pre = packed;
    const v4f* pim = packed + PACK_PER_MAT;

    for (int nt = wave; nt < NTILES; nt += WAVES_WG) {
        const v4f* pre_nt = pre + (long)nt * PACK_PER_NT + lane;
        const v4f* pim_nt = pim + (long)nt * PACK_PER_NT + lane;

        v8f acc_re0 = {}, acc_im0 = {}, acc_re1 = {}, acc_im1 = {};

        for (int q0 = 0; q0 < QGROUPS; q0 += 2) {          // K step = 16
            if (q0 + 8 < QGROUPS) {                        // 4KB ahead
                __builtin_prefetch(pre_nt + (q0 + 8) * 32, 0, 1);
                __builtin_prefetch(pim_nt + (q0 + 8) * 32, 0, 1);
            }
            // Coalesced fragment loads: 512B contiguous per instruction.
            const v4f bre_a = pre_nt[(long)q0 * 32];
            const v4f bre_b = pre_nt[(long)(q0 + 1) * 32];
            const v4f bim_a = pim_nt[(long)q0 * 32];
            const v4f bim_b = pim_nt[(long)(q0 + 1) * 32];

            const v2f bres[4] = {
                __builtin_shufflevector(bre_a, bre_a, 0, 1),
                __builtin_shufflevector(bre_a, bre_a, 2, 3),
                __builtin_shufflevector(bre_b, bre_b, 0, 1),
                __builtin_shufflevector(bre_b, bre_b, 2, 3)};
            const v2f bims[4] = {
                __builtin_shufflevector(bim_a, bim_a, 0, 1),
                __builtin_shufflevector(bim_a, bim_a, 2, 3),
                __builtin_shufflevector(bim_b, bim_b, 0, 1),
                __builtin_shufflevector(bim_b, bim_b, 2, 3)};

#pragma unroll
            for (int j = 0; j < 4; ++j) {
                const int k = 8 * q0 + koff + 4 * j;
                v2f a0 = *(const v2f*)&lds_x[(row)      * HOP + k];
                v2f a1 = *(const v2f*)&lds_x[(16 + row) * HOP + k];
                acc_re0 = __builtin_amdgcn_wmma_f32_16x16x4_f32(
                    false, a0, false, bres[j], (short)0, acc_re0, false, false);
                acc_im0 = __builtin_amdgcn_wmma_f32_16x16x4_f32(
                    false, a0, false, bims[j], (short)0, acc_im0, false, false);
                acc_re1 = __builtin_amdgcn_wmma_f32_16x16x4_f32(
                    false, a1, false, bres[j], (short)0, acc_re1, false, false);
                acc_im1 = __builtin_amdgcn_wmma_f32_16x16x4_f32(
                    false, a1, false, bims[j], (short)0, acc_im1, false, false);
            }
        }
        store_tiles(out, OUT_HALF, b, t0, nt * 16 + row, hi,
                    acc_re0, acc_im0, acc_re1, acc_im1);
    }
}

// ---------------------------------------------------------------------------
// Fallback: direct (unpacked) weight loads — used only if d_ws is too small.
// ---------------------------------------------------------------------------
__global__ __launch_bounds__(THREADS, 1)
void stft_wmma_f32_direct(const float* __restrict__ x,
                          const float* __restrict__ wre,
                          const float* __restrict__ wim,
                          float* __restrict__ out) {
    extern __shared__ float lds_x[];

    const int wg   = blockIdx.x;
    const int b    = wg / BLK_PER_B;
    const int t0   = (wg % BLK_PER_B) * FRAMES_WG;
    const long j0  = (long)t0 * HOP - PAD;

    const float* xb = x + (long)b * N_SAMP;
    for (int i = threadIdx.x; i < LDS_FLOATS; i += THREADS) {
        long j = j0 + i;
        if (j < 0)            j = -j;
        else if (j >= N_SAMP) j = 2L * (N_SAMP - 1) - j;
        lds_x[i] = xb[j];
    }
    __syncthreads();

    const int lane = threadIdx.x & 31;
    const int wave = threadIdx.x >> 5;
    const int row  = lane & 15;
    const int hi   = lane >> 4;
    const int koff = hi << 1;
    const long OUT_HALF = (long)N_BATCH * T_FRAMES * N_FREQ;

    for (int nt = wave; nt < NTILES; nt += WAVES_WG) {
        const int f  = nt * 16 + row;
        const int fc = (f < N_FREQ) ? f : (N_FREQ - 1);
        const float* wre_p = wre + (long)fc * N_FFT;
        const float* wim_p = wim + (long)fc * N_FFT;

        v8f acc_re0 = {}, acc_im0 = {}, acc_re1 = {}, acc_im1 = {};

        for (int k0 = 0; k0 < N_FFT; k0 += 16) {
#pragma unroll
            for (int j = 0; j < 4; ++j) {
                const int k = k0 + 4 * j + koff;
                v2f a0  = *(const v2f*)&lds_x[(row)      * HOP + k];
                v2f a1  = *(const v2f*)&lds_x[(16 + row) * HOP + k];
                v2f bre = *(const v2f*)&wre_p[k];
                v2f bim = *(const v2f*)&wim_p[k];
                acc_re0 = __builtin_amdgcn_wmma_f32_16x16x4_f32(
                    false, a0, false, bre, (short)0, acc_re0, false, false);
                acc_im0 = __builtin_amdgcn_wmma_f32_16x16x4_f32(
                    false, a0, false, bim, (short)0, acc_im0, false, false);
                acc_re1 = __builtin_amdgcn_wmma_f32_16x16x4_f32(
                    false, a1, false, bre, (short)0, acc_re1, false, false);
                acc_im1 = __builtin_amdgcn_wmma_f32_16x16x4_f32(
                    false, a1, false, bim, (short)0, acc_im1, false, false);
            }
        }
        store_tiles(out, OUT_HALF, b, t0, f, hi,
                    acc_re0, acc_im0, acc_re1, acc_im1);
    }
}

extern "C" void kernel_launch(void* const* d_in, const int* in_sizes, int n_in,
                              void* d_out, int out_size, void* d_ws, size_t ws_size,
                              hipStream_t stream) {
    const float* x   = (const float*)d_in[0];
    const float* wre = (const float*)d_in[1];
    const float* wim = (const float*)d_in[2];
    float* out = (float*)d_out;

    const size_t lds_bytes = (size_t)LDS_FLOATS * sizeof(float); // 71680 B
    (void)hipFuncSetAttribute((const void*)stft_wmma_f32_packed,
                              hipFuncAttributeMaxDynamicSharedMemorySize,
                              (int)lds_bytes);
    (void)hipFuncSetAttribute((const void*)stft_wmma_f32_direct,
                              hipFuncAttributeMaxDynamicSharedMemorySize,
                              (int)lds_bytes);

    const dim3 grid(N_BATCH * BLK_PER_B);   // 432 workgroups

    if (ws_size >= PACK_BYTES) {
        const int total  = 2 * PACK_PER_MAT;
        const int blocks = (total + 255) / 256;
        stft_repack_weights<<<blocks, 256, 0, stream>>>(wre, wim, (v4f*)d_ws);
        stft_wmma_f32_packed<<<grid, THREADS, lds_bytes, stream>>>(
            x, (const v4f*)d_ws, out);
    } else {
        stft_wmma_f32_direct<<<grid, THREADS, lds_bytes, stream>>>(
            x, wre, wim, out);
    }
}